// PurposivePrunerLoss_2757369004417
// MI455X (gfx1250) — compile-verified
//
#include <hip/hip_runtime.h>
#include <float.h>
#include <math.h>

// ---------------------------------------------------------------------------
// Masked logsumexp loss ("PurposivePrunerLoss"):
//   keep logit0 always; keep logit j (j>=1) iff cos[j-1] < 0.8
//   loss_b = logsumexp(kept) - logit0 ; result = mean_b(loss_b)
// Memory-bound streaming reduction: 128 MiB @ 23.3 TB/s ~ 5.6 us.
// Strategy: async global->LDS staging (CDNA5 ASYNCcnt path), two passes over
// the LDS tile (max, then sum of exp), branchless -FLT_MAX masking.
// ---------------------------------------------------------------------------

#define SIM_THR 0.8f

constexpr int TPB    = 256;   // threads per block (8 wave32s)
constexpr int SPLIT  = 8;     // chunks per row
constexpr int TILE   = 8;     // elements per thread per staged tile
constexpr int NWAVES = TPB / 32;

#define GAS __attribute__((address_space(1)))
#define LAS __attribute__((address_space(3)))

#if defined(__gfx1250__) && \
    __has_builtin(__builtin_amdgcn_global_load_async_to_lds_b32) && \
    __has_builtin(__builtin_amdgcn_s_wait_asynccnt)
#define ASYNC_PATH 1
#else
#define ASYNC_PATH 0
#endif

__device__ __forceinline__ void lse_combine(float& m, float& s, float om, float os) {
    // merge two (max, sum-of-exp-rebased) pairs; -FLT_MAX acts as "empty"
    float M = fmaxf(m, om);
    s = s * __expf(m - M) + os * __expf(om - M);
    m = M;
}

// ---------------------------------------------------------------------------
// Kernel A: per-(row,chunk) partial (m, s) written to workspace.
// blockIdx.x = row * SPLIT + chunk
// ---------------------------------------------------------------------------
__global__ __launch_bounds__(TPB)
void lse_partial_kernel(const float* __restrict__ outp,  // [B, K+1]
                        const float* __restrict__ simi,  // [B, K]
                        float2* __restrict__ part,       // [B*SPLIT]
                        int K, int rowStride, int ntiles) {
    const int tid   = threadIdx.x;
    const int row   = blockIdx.x / SPLIT;
    const int chunk = blockIdx.x % SPLIT;
    const size_t rowO = (size_t)row * (size_t)rowStride; // logits row base
    const size_t rowS = (size_t)row * (size_t)K;         // cos-sim row base
    const int chunkBase = chunk * ntiles * (TPB * TILE);

    float m = -FLT_MAX;
    float s = 0.0f;

#if ASYNC_PATH
    __shared__ float sLog[2][TILE * TPB];
    __shared__ float sSim[2][TILE * TPB];

    auto issue_tile = [&](int t, int buf) {
        const int base = chunkBase + t * (TPB * TILE);
#pragma unroll
        for (int k = 0; k < TILE; ++k) {
            int j  = base + k * TPB + tid;
            int jc = (j < K) ? j : 0;  // clamp address; value masked later
            __builtin_amdgcn_global_load_async_to_lds_b32(
                (GAS int*)(simi + rowS + jc),
                (LAS int*)&sSim[buf][k * TPB + tid], 0, 0);
            __builtin_amdgcn_global_load_async_to_lds_b32(
                (GAS int*)(outp + rowO + 1 + jc),
                (LAS int*)&sLog[buf][k * TPB + tid], 0, 0);
        }
    };

    issue_tile(0, 0);
    if (ntiles > 1) issue_tile(1, 1);

    for (int t = 0; t < ntiles; ++t) {
        // each tile = 2*TILE async b32 ops in flight per lane (in-order done)
        if (t + 1 < ntiles) __builtin_amdgcn_s_wait_asynccnt(2 * TILE);
        else                __builtin_amdgcn_s_wait_asynccnt(0);
        asm volatile("" ::: "memory");

        const int buf  = t & 1;
        const int base = chunkBase + t * (TPB * TILE);
        float v[TILE];
        float mt = -FLT_MAX;
#pragma unroll
        for (int k = 0; k < TILE; ++k) {
            int   j  = base + k * TPB + tid;
            float lg = sLog[buf][k * TPB + tid];
            float cs = sSim[buf][k * TPB + tid];
            float ve = ((j < K) && (cs < SIM_THR)) ? lg : -FLT_MAX;
            v[k] = ve;
            mt   = fmaxf(mt, ve);
        }
        float M = fmaxf(m, mt);
        s *= __expf(m - M);           // rescale once per tile
#pragma unroll
        for (int k = 0; k < TILE; ++k) s += __expf(v[k] - M);
        m = M;

        asm volatile("" ::: "memory");
        if (t + 2 < ntiles) issue_tile(t + 2, buf);
    }
#else
    // Fallback: direct coalesced b32 loads, identical math.
    for (int t = 0; t < ntiles; ++t) {
        const int base = chunkBase + t * (TPB * TILE);
        float v[TILE];
        float mt = -FLT_MAX;
#pragma unroll
        for (int k = 0; k < TILE; ++k) {
            int   j  = base + k * TPB + tid;
            int   jc = (j < K) ? j : 0;
            float lg = outp[rowO + 1 + jc];
            float cs = simi[rowS + jc];
            float ve = ((j < K) && (cs < SIM_THR)) ? lg : -FLT_MAX;
            v[k] = ve;
            mt   = fmaxf(mt, ve);
        }
        float M = fmaxf(m, mt);
        s *= __expf(m - M);
#pragma unroll
        for (int k = 0; k < TILE; ++k) s += __expf(v[k] - M);
        m = M;
    }
#endif

    // wave32 shuffle reduction
    for (int off = 16; off >= 1; off >>= 1) {
        float om = __shfl_xor(m, off, 32);
        float os = __shfl_xor(s, off, 32);
        lse_combine(m, s, om, os);
    }
    __shared__ float rM[NWAVES];
    __shared__ float rS[NWAVES];
    const int wid  = tid >> 5;
    const int lane = tid & 31;
    if (lane == 0) { rM[wid] = m; rS[wid] = s; }
    __syncthreads();
    if (tid == 0) {
        float M = rM[0], S = rS[0];
        for (int w = 1; w < NWAVES; ++w) lse_combine(M, S, rM[w], rS[w]);
        part[blockIdx.x] = make_float2(M, S);
    }
}

// ---------------------------------------------------------------------------
// Kernel B: combine SPLIT partials per row, inject logit0, mean over rows.
// Single block -> fully deterministic reduction order.
// ---------------------------------------------------------------------------
__global__ __launch_bounds__(TPB)
void lse_final_kernel(const float* __restrict__ outp,
                      const float2* __restrict__ part,
                      float* __restrict__ res,
                      int B, int rowStride) {
    const int tid = threadIdx.x;
    float acc = 0.0f;
    for (int r = tid; r < B; r += TPB) {
        const float logit0 = outp[(size_t)r * (size_t)rowStride];
        float M = logit0, S = 1.0f;   // positive logit always kept
        for (int c = 0; c < SPLIT; ++c) {
            float2 p = part[r * SPLIT + c];
            lse_combine(M, S, p.x, p.y);
        }
        acc += M + logf(S) - logit0;
    }
    for (int off = 16; off >= 1; off >>= 1)
        acc += __shfl_xor(acc, off, 32);
    __shared__ float rAcc[NWAVES];
    const int wid  = tid >> 5;
    const int lane = tid & 31;
    if (lane == 0) rAcc[wid] = acc;
    __syncthreads();
    if (tid == 0) {
        float t = 0.0f;
        for (int w = 0; w < NWAVES; ++w) t += rAcc[w];
        res[0] = t / (float)B;
    }
}

// ---------------------------------------------------------------------------
extern "C" void kernel_launch(void* const* d_in, const int* in_sizes, int n_in,
                              void* d_out, int out_size, void* d_ws, size_t ws_size,
                              hipStream_t stream) {
    const float* outp = (const float*)d_in[0];       // [B, K+1] f32
    const float* simi = (const float*)d_in[1];       // [B, K]   f32
    (void)n_in; (void)out_size; (void)ws_size;

    const int B         = in_sizes[2];               // target: [B]
    const int K         = in_sizes[1] / B;           // 65536
    const int rowStride = in_sizes[0] / B;           // K + 1

    const int tileElems = TPB * TILE;                                  // 2048
    const int perChunk  = (K + SPLIT - 1) / SPLIT;                     // 8192
    const int ntiles    = (perChunk + tileElems - 1) / tileElems;      // 4

    float2* part = (float2*)d_ws;                    // B*SPLIT float2 = 16 KB

    lse_partial_kernel<<<B * SPLIT, TPB, 0, stream>>>(outp, simi, part,
                                                      K, rowStride, ntiles);
    lse_final_kernel<<<1, TPB, 0, stream>>>(outp, part, (float*)d_out,
                                            B, rowStride);
}